// HashGridEncoder_17617955848983
// MI455X (gfx1250) — compile-verified
//
#include <hip/hip_runtime.h>
#include <hip/hip_bf16.h>

// ---------------------------------------------------------------------------
// Instant-NGP hash-grid encoder for MI455X (gfx1250).
// One workgroup per (level, 8192-point chunk). Stage the level's feature
// table (<=256KB) into the 320KB CDNA5 LDS via global_load_async_to_lds_b128,
// wait on ASYNCcnt, then do all 8-corner gathers as ds_load_b64 from LDS
// instead of random global loads (L2-line-granular) -- converts 128M random
// 8B VMEM gathers into LDS-rate gathers; streaming HBM traffic is just
// x (12MB) + out (128MB) + table (4MB, L2-resident).
// ---------------------------------------------------------------------------

#define N_LEVELS   16
#define LOG2_T     15
#define TBL        (1 << LOG2_T)          // 32768 entries per level
#define TMASK      (TBL - 1)
#define F_PER_LVL  2
#define THREADS    512
#define PPB        8192                   // points per block

#define PRIME_Y    2654435761u
#define PRIME_Z    805459861u

// RESOLUTIONS[l] = ceil(16 * (32)^(l/15)) + 1
__constant__ int c_res[N_LEVELS] = {
    17, 22, 27, 33, 42, 52, 65, 82, 103, 129, 163, 205, 257, 324, 408, 513
};
// entries actually used per level: dense levels (res^3 <= T) use res^3 rows
__constant__ int c_entries[N_LEVELS] = {
    4913, 10648, 19683, TBL, TBL, TBL, TBL, TBL,
    TBL,  TBL,   TBL,   TBL, TBL, TBL, TBL, TBL
};

#if defined(__AMDGCN__) && \
    __has_builtin(__builtin_amdgcn_global_load_async_to_lds_b128) && \
    __has_builtin(__builtin_amdgcn_s_wait_asynccnt)
#define HG_ASYNC 1
#else
#define HG_ASYNC 0
#endif

// Pointer types matching the async-load builtin signature:
//   (v4i AS1* gsrc, v4i AS3* ldst, imm offset, imm cpol)
typedef int v4i __attribute__((vector_size(16)));
typedef __attribute__((address_space(1))) v4i* gptr_v4i;
typedef __attribute__((address_space(3))) v4i* lptr_v4i;

__global__ __launch_bounds__(THREADS, 1)
void HashGridEncoder_kernel(const float* __restrict__ x,
                            const float* __restrict__ table,
                            float* __restrict__ out,
                            int npoints)
{
    // 256 KB LDS: full hashed-level table. CDNA5 WGP has 320 KB LDS.
    __shared__ float4 s_tbl4[TBL * F_PER_LVL / 4];   // 16384 x float4
    float2* s_tbl = (float2*)s_tbl4;

    const int l       = blockIdx.y;
    const int res     = c_res[l];
    const int entries = c_entries[l];
    const bool dense  = (entries < TBL) || (res * res * res <= TBL);

    // ---- Stage this level's table slice into LDS ----
    const float4* g4 = (const float4*)(table + (size_t)l * (TBL * F_PER_LVL));
    const int nvec = (entries * F_PER_LVL + 3) >> 2;   // ragged tail reads stay
                                                       // inside the 256KB slice
#if HG_ASYNC
    for (int v = threadIdx.x; v < nvec; v += THREADS) {
        __builtin_amdgcn_global_load_async_to_lds_b128(
            (gptr_v4i)(g4 + v),
            (lptr_v4i)(s_tbl4 + v),
            0, 0);
    }
    __builtin_amdgcn_s_wait_asynccnt(0);
#else
    for (int v = threadIdx.x; v < nvec; v += THREADS) {
        s_tbl4[v] = g4[v];
    }
#endif
    __syncthreads();

    // ---- Gather + trilinear interpolation ----
    const int base = blockIdx.x * PPB;
    const float resm1 = (float)(res - 1);
    const int res2 = res * res;

    for (int i = threadIdx.x; i < PPB; i += THREADS) {
        const int n = base + i;
        if (n >= npoints) break;

        const float px = x[(size_t)n * 3 + 0];
        const float py = x[(size_t)n * 3 + 1];
        const float pz = x[(size_t)n * 3 + 2];

        // Match reference rounding: u=(x+1)/2, then pos=u*(res-1)
        const float fx = ((px + 1.0f) * 0.5f) * resm1;
        const float fy = ((py + 1.0f) * 0.5f) * resm1;
        const float fz = ((pz + 1.0f) * 0.5f) * resm1;

        const float gx = floorf(fx), gy = floorf(fy), gz = floorf(fz);
        const float wx = fx - gx, wy = fy - gy, wz = fz - gz;

        int ix = (int)gx; ix = ix < 0 ? 0 : (ix > res - 2 ? res - 2 : ix);
        int iy = (int)gy; iy = iy < 0 ? 0 : (iy > res - 2 ? res - 2 : iy);
        int iz = (int)gz; iz = iz < 0 ? 0 : (iz > res - 2 ? res - 2 : iz);

        const float wxv[2] = {1.0f - wx, wx};
        const float wyv[2] = {1.0f - wy, wy};
        const float wzv[2] = {1.0f - wz, wz};

        float acc0 = 0.0f, acc1 = 0.0f;

        if (dense) {
            const int bx[2] = { ix,            ix + 1 };
            const int by[2] = { iy * res,      (iy + 1) * res };
            const int bz[2] = { iz * res2,     (iz + 1) * res2 };
#pragma unroll
            for (int c = 0; c < 8; ++c) {
                const int a = (c >> 2) & 1, b = (c >> 1) & 1, d = c & 1;
                const int idx = bx[a] + by[b] + bz[d];
                const float2 f = s_tbl[idx];
                const float w = wxv[a] * wyv[b] * wzv[d];
                acc0 = fmaf(w, f.x, acc0);
                acc1 = fmaf(w, f.y, acc1);
            }
        } else {
            const unsigned hx[2] = { (unsigned)ix,
                                     (unsigned)(ix + 1) };
            const unsigned hy[2] = { (unsigned)iy * PRIME_Y,
                                     (unsigned)(iy + 1) * PRIME_Y };
            const unsigned hz[2] = { (unsigned)iz * PRIME_Z,
                                     (unsigned)(iz + 1) * PRIME_Z };
#pragma unroll
            for (int c = 0; c < 8; ++c) {
                const int a = (c >> 2) & 1, b = (c >> 1) & 1, d = c & 1;
                const int idx = (int)((hx[a] ^ hy[b] ^ hz[d]) & TMASK);
                const float2 f = s_tbl[idx];
                const float w = wxv[a] * wyv[b] * wzv[d];
                acc0 = fmaf(w, f.x, acc0);
                acc1 = fmaf(w, f.y, acc1);
            }
        }

        // out[n, 2l : 2l+2]
        float2* o = (float2*)(out + (size_t)n * (N_LEVELS * F_PER_LVL) + 2 * l);
        *o = make_float2(acc0, acc1);
    }
}

extern "C" void kernel_launch(void* const* d_in, const int* in_sizes, int n_in,
                              void* d_out, int out_size, void* d_ws, size_t ws_size,
                              hipStream_t stream) {
    const float* x     = (const float*)d_in[0];   // [N,3] f32
    const float* table = (const float*)d_in[1];   // [16,32768,2] f32
    float* out         = (float*)d_out;           // [N,32] f32
    const int npoints  = in_sizes[0] / 3;

    dim3 grid((npoints + PPB - 1) / PPB, N_LEVELS);
    HashGridEncoder_kernel<<<grid, THREADS, 0, stream>>>(x, table, out, npoints);
}